// STGCN_56831007261389
// MI455X (gfx1250) — compile-verified
//
#include <hip/hip_runtime.h>

typedef __attribute__((ext_vector_type(2))) float v2f;
typedef __attribute__((ext_vector_type(8))) float v8f;

__device__ __forceinline__ float sigmoidf_(float x) {
    return 1.0f / (1.0f + __expf(-x));
}

// ---------------------------------------------------------------------------
// Stage 1 conv: c_in = 1, KT = 3, GLU.  x:(N,12) -> out:(N,64,10)
// ---------------------------------------------------------------------------
__global__ void __launch_bounds__(256)
conv1_glu_kernel(const float* __restrict__ x, const float* __restrict__ W,
                 const float* __restrict__ b, float* __restrict__ out, int Nn)
{
    int idx = blockIdx.x * blockDim.x + threadIdx.x;
    int total = Nn * 64 * 10;
    if (idx >= total) return;
    int t = idx % 10;
    int tmp = idx / 10;
    int o = tmp % 64;
    int n = tmp / 64;
    const float* xr = x + n * 12 + t;
    float ya = b[o], yg = b[o + 64];
#pragma unroll
    for (int k = 0; k < 3; ++k) {
        float xv = xr[k];
        ya += xv * W[o * 3 + k];
        yg += xv * W[(o + 64) * 3 + k];
    }
    out[idx] = ya * sigmoidf_(yg);   // idx == n*640 + o*10 + t
}

// ---------------------------------------------------------------------------
// Generic temporal conv + GLU via V_WMMA_F32_16X16X4_F32.
// in : (N, c_in, T_in) row-major, W : (128, c_in*3) row-major, bias : (128)
// out: (N, 64, T_out) row-major (GLU output).  K = c_in*3, multiple of 4.
// One wave = one 16x16 'a' tile + matching 16x16 'g' tile over same columns.
// ---------------------------------------------------------------------------
__global__ void __launch_bounds__(256)
conv_glu_wmma_kernel(const float* __restrict__ in, const float* __restrict__ W,
                     const float* __restrict__ bias, float* __restrict__ out,
                     int Nn, int c_in, int T_in, int T_out)
{
    const int lane  = threadIdx.x & 31;
    const int wid   = threadIdx.x >> 5;
    const int K     = c_in * 3;
    const int ncols = Nn * T_out;          // multiple of 16 for all call sites
    const int col_tiles = ncols >> 4;

    int tile = blockIdx.x * 8 + wid;       // 4 row tiles * col_tiles total
    int col_tile = tile >> 2;
    int rt       = tile & 3;               // row tile 0..3 (a-channels 16*rt..)
    if (col_tile >= col_tiles) return;     // wave-uniform exit (EXEC all-1s kept)

    const int mrow  = lane & 15;           // A row / B col within tile
    const int khalf = lane >> 4;           // K sub-pair select (0 or 1)

    int col = col_tile * 16 + mrow;        // global column = (n, t)
    int n   = col / T_out;
    int t   = col - n * T_out;

    const float* Wa   = W + (rt * 16 + mrow) * K;        // a-channel weights
    const float* Wg   = W + (rt * 16 + mrow + 64) * K;   // gate weights
    const float* bcol = in + n * (c_in * T_in) + t;

    v8f acc_a = {};
    v8f acc_g = {};

    for (int kk = 0; kk < K; kk += 4) {
        int k0 = kk + 2 * khalf;
        int k1 = k0 + 1;
        v2f a, g, bf;
        a.x = Wa[k0];  a.y = Wa[k1];
        g.x = Wg[k0];  g.y = Wg[k1];
        int ci0 = k0 / 3, kq0 = k0 - 3 * ci0;
        int ci1 = k1 / 3, kq1 = k1 - 3 * ci1;
        bf.x = bcol[ci0 * T_in + kq0];     // im2col element B[k0][col]
        bf.y = bcol[ci1 * T_in + kq1];
        acc_a = __builtin_amdgcn_wmma_f32_16x16x4_f32(false, a, false, bf,
                                                      (short)0, acc_a, false, false);
        acc_g = __builtin_amdgcn_wmma_f32_16x16x4_f32(false, g, false, bf,
                                                      (short)0, acc_g, false, false);
    }

    float* obase = out + n * (64 * T_out) + t;
#pragma unroll
    for (int v = 0; v < 8; ++v) {
        int m = rt * 16 + v + 8 * khalf;   // output channel for this acc elem
        float av = acc_a[v] + bias[m];
        float gv = acc_g[v] + bias[m + 64];
        obase[m * T_out] = av * sigmoidf_(gv);
    }
}

// ---------------------------------------------------------------------------
// GCN dense transform: HW (M,16) = X (M,64) @ Wg (64,16) via f32 WMMA.
// M is an exact multiple of 16 (500000 / 300000).
// ---------------------------------------------------------------------------
__global__ void __launch_bounds__(256)
gcn_gemm_wmma_kernel(const float* __restrict__ X, const float* __restrict__ Wg,
                     float* __restrict__ HW, int M)
{
    const int lane = threadIdx.x & 31;
    const int wid  = threadIdx.x >> 5;
    int tile = blockIdx.x * 8 + wid;
    int row_tiles = M >> 4;
    if (tile >= row_tiles) return;         // wave-uniform

    const int mrow  = lane & 15;
    const int khalf = lane >> 4;
    const float* xr = X + (tile * 16 + mrow) * 64;

    v8f acc = {};
    for (int kk = 0; kk < 64; kk += 4) {
        int k0 = kk + 2 * khalf;
        v2f a, b;
        a.x = xr[k0];
        a.y = xr[k0 + 1];
        b.x = Wg[k0 * 16 + mrow];
        b.y = Wg[(k0 + 1) * 16 + mrow];
        acc = __builtin_amdgcn_wmma_f32_16x16x4_f32(false, a, false, b,
                                                    (short)0, acc, false, false);
    }
#pragma unroll
    for (int v = 0; v < 8; ++v) {
        int mr = tile * 16 + v + 8 * khalf;
        HW[mr * 16 + mrow] = acc[v];
    }
}

// ---------------------------------------------------------------------------
// GCN aggregation pieces
// ---------------------------------------------------------------------------
__global__ void __launch_bounds__(256)
deg_init_kernel(float* __restrict__ deg, int M)
{
    int i = blockIdx.x * blockDim.x + threadIdx.x;
    if (i < M) deg[i] = 1.0f;              // self loop
}

__global__ void __launch_bounds__(256)
deg_count_kernel(float* __restrict__ deg, const int* __restrict__ dst, int E)
{
    int e = blockIdx.x * blockDim.x + threadIdx.x;
    if (e < E) atomicAdd(&deg[dst[e]], 1.0f);
}

__global__ void __launch_bounds__(256)
dinv_kernel(float* __restrict__ deg, int M)
{
    int i = blockIdx.x * blockDim.x + threadIdx.x;
    if (i < M) deg[i] = rsqrtf(deg[i]);    // deg >= 1 always (self loop)
}

// out[m][c] = bias[c] + HW[m][c]*dinv[m]^2   (self-loop message + bias)
__global__ void __launch_bounds__(256)
gcn_init_kernel(float* __restrict__ outp, const float* __restrict__ HW,
                const float* __restrict__ dinv, const float* __restrict__ bg, int M)
{
    int idx = blockIdx.x * blockDim.x + threadIdx.x;
    if (idx >= M * 16) return;
    int m = idx >> 4, c = idx & 15;
    float di = dinv[m];
    outp[idx] = bg[c] + HW[idx] * di * di;
}

// 16 consecutive lanes per edge -> coalesced gather of HW[src], atomic scatter
__global__ void __launch_bounds__(256)
gcn_scatter_kernel(float* __restrict__ outp, const float* __restrict__ HW,
                   const float* __restrict__ dinv, const int* __restrict__ src,
                   const int* __restrict__ dst, int E)
{
    int idx = blockIdx.x * blockDim.x + threadIdx.x;
    if (idx >= E * 16) return;
    int e = idx >> 4, c = idx & 15;
    int s = src[e], d = dst[e];
    float w = dinv[s] * dinv[d];
    atomicAdd(&outp[d * 16 + c], HW[s * 16 + c] * w);
}

__global__ void __launch_bounds__(256)
relu_kernel(float* __restrict__ p, int total)
{
    int i = blockIdx.x * blockDim.x + threadIdx.x;
    if (i < total) p[i] = fmaxf(p[i], 0.0f);
}

// ---------------------------------------------------------------------------
// LayerNorm over (64, T) per sample; optional (0,2,1) transpose on output.
// One wave32 per sample; shuffle reduction.
// ---------------------------------------------------------------------------
__global__ void __launch_bounds__(256)
layernorm_kernel(const float* __restrict__ in, const float* __restrict__ g,
                 const float* __restrict__ be, float* __restrict__ out,
                 int Nn, int T, int transpose_out)
{
    int lane = threadIdx.x & 31;
    int wid  = threadIdx.x >> 5;
    int n = blockIdx.x * 8 + wid;
    if (n >= Nn) return;                   // wave-uniform
    int len = 64 * T;
    const float* xr = in + n * len;
    float s = 0.0f, sq = 0.0f;
    for (int e = lane; e < len; e += 32) {
        float v = xr[e];
        s += v; sq += v * v;
    }
#pragma unroll
    for (int m = 16; m >= 1; m >>= 1) {
        s  += __shfl_xor(s,  m, 32);
        sq += __shfl_xor(sq, m, 32);
    }
    float mean = s / (float)len;
    float var  = sq / (float)len - mean * mean;   // biased, matches jnp.var
    float inv  = rsqrtf(var + 1e-5f);
    float* orow = out + n * len;
    for (int e = lane; e < len; e += 32) {
        float y = (xr[e] - mean) * inv * g[e] + be[e];   // g/be are (64,T) flat
        if (transpose_out) {
            int c = e / T, t = e - c * T;
            orow[t * 64 + c] = y;
        } else {
            orow[e] = y;
        }
    }
}

// ---------------------------------------------------------------------------
// Output head: h5 (N,64,2) -> Linear(2,1) -> Linear(64,1) -> out (N,1)
// ---------------------------------------------------------------------------
__global__ void __launch_bounds__(256)
output_final_kernel(const float* __restrict__ h5, const float* __restrict__ Wlin,
                    const float* __restrict__ blin, const float* __restrict__ Wfc,
                    const float* __restrict__ bfc, float* __restrict__ out, int Nn)
{
    int n = blockIdx.x * blockDim.x + threadIdx.x;
    if (n >= Nn) return;
    float w0 = Wlin[0], w1 = Wlin[1], bl = blin[0];
    float acc = bfc[0];
    const float* hr = h5 + n * 128;
#pragma unroll
    for (int c = 0; c < 64; ++c)
        acc += Wfc[c] * (bl + w0 * hr[c * 2] + w1 * hr[c * 2 + 1]);
    out[n] = acc;
}

// ---------------------------------------------------------------------------
static inline int cdiv(long a, int b) { return (int)((a + b - 1) / b); }

extern "C" void kernel_launch(void* const* d_in, const int* in_sizes, int n_in,
                              void* d_out, int out_size, void* d_ws, size_t ws_size,
                              hipStream_t stream)
{
    const float* x    = (const float*)d_in[0];
    const int*   ei   = (const int*)  d_in[1];
    // d_in[2] = num_edges (device scalar) — recover NE on host from sizes:
    const int NE = in_sizes[1] / 24;          // edge_index is (2, 12*NE)
    const int Nn = in_sizes[0] / 12;          // x is (N, 12)

    const float* Wt1a = (const float*)d_in[3];  const float* bt1a = (const float*)d_in[4];
    const float* Wg1  = (const float*)d_in[5];  const float* bg1  = (const float*)d_in[6];
    const float* Wt1b = (const float*)d_in[7];  const float* bt1b = (const float*)d_in[8];
    const float* g1   = (const float*)d_in[9];  const float* be1  = (const float*)d_in[10];
    const float* Wt2a = (const float*)d_in[11]; const float* bt2a = (const float*)d_in[12];
    const float* Wg2  = (const float*)d_in[13]; const float* bg2  = (const float*)d_in[14];
    const float* Wt2b = (const float*)d_in[15]; const float* bt2b = (const float*)d_in[16];
    const float* g2   = (const float*)d_in[17]; const float* be2  = (const float*)d_in[18];
    const float* Wto  = (const float*)d_in[19]; const float* bto  = (const float*)d_in[20];
    const float* Wlin = (const float*)d_in[21]; const float* blin = (const float*)d_in[22];
    const float* Wfc  = (const float*)d_in[23]; const float* bfc  = (const float*)d_in[24];

    const int* srcE = ei;                 // row 0 of edge_index
    const int* dstE = ei + (size_t)12 * NE;  // row 1

    // Workspace layout (reused regions; lifetimes disjoint):
    //  A: h1 (N*640) -> h2t (N*512) -> h4 (N*256)
    //  B: h2 (N*512) -> h3 (N*384)  -> h5 (N*128)
    //  C: gcn out (M1*16 = N*160)   -> (M2*16 = N*96)
    //  D: HW (N*160) -> (N*96)
    //  E: deg/dinv (N*10)
    char* ws = (char*)d_ws;
    size_t off = 0;
    float* bufA = (float*)(ws + off); off += (size_t)Nn * 640 * sizeof(float);
    float* bufB = (float*)(ws + off); off += (size_t)Nn * 512 * sizeof(float);
    float* bufC = (float*)(ws + off); off += (size_t)Nn * 160 * sizeof(float);
    float* bufD = (float*)(ws + off); off += (size_t)Nn * 160 * sizeof(float);
    float* deg  = (float*)(ws + off); off += (size_t)Nn * 10  * sizeof(float);
    (void)ws_size; (void)n_in; (void)out_size;

    const int M1 = Nn * 10, E1 = 10 * NE;
    const int M2 = Nn * 6,  E2 = 6  * NE;

    // ---- block 1 ----
    conv1_glu_kernel<<<cdiv((long)Nn * 640, 256), 256, 0, stream>>>(x, Wt1a, bt1a, bufA, Nn);

    gcn_gemm_wmma_kernel<<<cdiv((long)(M1 >> 4), 8), 256, 0, stream>>>(bufA, Wg1, bufD, M1);
    deg_init_kernel <<<cdiv(M1, 256), 256, 0, stream>>>(deg, M1);
    deg_count_kernel<<<cdiv(E1, 256), 256, 0, stream>>>(deg, dstE, E1);
    dinv_kernel     <<<cdiv(M1, 256), 256, 0, stream>>>(deg, M1);
    gcn_init_kernel <<<cdiv((long)M1 * 16, 256), 256, 0, stream>>>(bufC, bufD, deg, bg1, M1);
    gcn_scatter_kernel<<<cdiv((long)E1 * 16, 256), 256, 0, stream>>>(bufC, bufD, deg, srcE, dstE, E1);
    relu_kernel     <<<cdiv((long)M1 * 16, 256), 256, 0, stream>>>(bufC, M1 * 16);

    // (N,16,10) -> conv GLU -> h2 (N,64,8)
    conv_glu_wmma_kernel<<<cdiv((long)(Nn * 8 / 16) * 4, 8), 256, 0, stream>>>(
        bufC, Wt1b, bt1b, bufB, Nn, 16, 10, 8);
    // layernorm + fused transpose(0,2,1): h2 (B) -> h2t (A), viewed as (N,64,8)
    layernorm_kernel<<<cdiv(Nn, 8), 256, 0, stream>>>(bufB, g1, be1, bufA, Nn, 8, 1);

    // ---- block 2 ----
    conv_glu_wmma_kernel<<<cdiv((long)(Nn * 6 / 16) * 4, 8), 256, 0, stream>>>(
        bufA, Wt2a, bt2a, bufB, Nn, 64, 8, 6);

    gcn_gemm_wmma_kernel<<<cdiv((long)(M2 >> 4), 8), 256, 0, stream>>>(bufB, Wg2, bufD, M2);
    deg_init_kernel <<<cdiv(M2, 256), 256, 0, stream>>>(deg, M2);
    deg_count_kernel<<<cdiv(E2, 256), 256, 0, stream>>>(deg, dstE, E2);
    dinv_kernel     <<<cdiv(M2, 256), 256, 0, stream>>>(deg, M2);
    gcn_init_kernel <<<cdiv((long)M2 * 16, 256), 256, 0, stream>>>(bufC, bufD, deg, bg2, M2);
    gcn_scatter_kernel<<<cdiv((long)E2 * 16, 256), 256, 0, stream>>>(bufC, bufD, deg, srcE, dstE, E2);
    relu_kernel     <<<cdiv((long)M2 * 16, 256), 256, 0, stream>>>(bufC, M2 * 16);

    // (N,16,6) -> conv GLU -> h4 (N,64,4), layernorm in place (no transpose)
    conv_glu_wmma_kernel<<<cdiv((long)(Nn * 4 / 16) * 4, 8), 256, 0, stream>>>(
        bufC, Wt2b, bt2b, bufA, Nn, 16, 6, 4);
    layernorm_kernel<<<cdiv(Nn, 8), 256, 0, stream>>>(bufA, g2, be2, bufA, Nn, 4, 0);

    // ---- output layer ----
    conv_glu_wmma_kernel<<<cdiv((long)(Nn * 2 / 16) * 4, 8), 256, 0, stream>>>(
        bufA, Wto, bto, bufB, Nn, 64, 4, 2);
    output_final_kernel<<<cdiv(Nn, 256), 256, 0, stream>>>(
        bufB, Wlin, blin, Wfc, bfc, (float*)d_out, Nn);
}